// EuclideanLoss_61280593379514
// MI455X (gfx1250) — compile-verified
//
#include <hip/hip_runtime.h>

// MI455X / gfx1250, wave32. Memory-bound kernel (1 GiB @ 23.3 TB/s ~= 46 us floor).
// Row-sum reduction is pushed into the matrix pipe via V_WMMA_F32_16X16X4_F32
// with an all-ones B matrix: C accumulates per-row sums of diff^2.

typedef float v2f __attribute__((ext_vector_type(2)));
typedef float v8f __attribute__((ext_vector_type(8)));

#define D_DIM 128   // embedding dim (fixed by reference)

// ---------------------------------------------------------------------------
// Kernel 1: inclusive prefix scan of segment lengths -> row offsets (nseg+1),
// and zero-init the output accumulator (harness poisons d_out with 0xAA).
// Single block of 256 threads, ~2048 segments: trivial cost.
// ---------------------------------------------------------------------------
__global__ void __launch_bounds__(256)
seg_scan_kernel(const int* __restrict__ num, int nseg,
                int* __restrict__ off, float* __restrict__ out)
{
    __shared__ int partials[256];
    const int tid = threadIdx.x;
    const int per = (nseg + 255) >> 8;           // segments per thread
    int local[32];
    int run = 0;
    for (int i = 0; i < per && i < 32; ++i) {
        int idx = tid * per + i;
        int v = (idx < nseg) ? num[idx] : 0;
        run += v;
        local[i] = run;                          // inclusive within chunk
    }
    partials[tid] = run;
    __syncthreads();
    // naive log-step inclusive scan over chunk totals
    for (int ofs = 1; ofs < 256; ofs <<= 1) {
        int add = (tid >= ofs) ? partials[tid - ofs] : 0;
        __syncthreads();
        partials[tid] += add;
        __syncthreads();
    }
    const int prev = (tid > 0) ? partials[tid - 1] : 0;
    for (int i = 0; i < per && i < 32; ++i) {
        int idx = tid * per + i;
        if (idx < nseg) off[idx + 1] = prev + local[i];
    }
    if (tid == 0) { off[0] = 0; *out = 0.0f; }
}

// ---------------------------------------------------------------------------
// Kernel 2: one wave per 16-row tile (grid-stride). Per K-chunk of 4:
// lane L loads float2 of row (L&15) at K = k0 + 2*(L>>4), forms diff^2 into
// the documented 16x4 f32 A-fragment layout, and WMMA-accumulates against a
// ones B. Two C accumulators break the WMMA->WMMA RAW chain.
// Finalize: lane 0 owns rows 0..7 (c[0..7]), lane 16 owns rows 8..15.
// ---------------------------------------------------------------------------
__global__ void __launch_bounds__(256)
euclid_loss_kernel(const float* __restrict__ a, const float* __restrict__ b,
                   const int* __restrict__ off, int nseg, int ntiles,
                   float* __restrict__ out)
{
    const int lane   = threadIdx.x & 31;
    const int gwave  = blockIdx.x * (blockDim.x >> 5) + (threadIdx.x >> 5);
    const int nwaves = gridDim.x * (blockDim.x >> 5);

    const int m  = lane & 15;          // tile row this lane supplies to A
    const int kp = (lane >> 4) << 1;   // K-pair offset within chunk: 0 or 2

    const v2f ones = {1.0f, 1.0f};
    float acc = 0.0f;

    for (int t = gwave; t < ntiles; t += nwaves) {
        const long rowbase = (long)t * 16;
        const float* pa = a + (rowbase + m) * D_DIM + kp;
        const float* pb = b + (rowbase + m) * D_DIM + kp;

        // Prefetch this lane's row for the next grid-stride tile.
        if (t + nwaves < ntiles) {
            const long nrow = (long)(t + nwaves) * 16 + m;
            __builtin_prefetch(a + nrow * D_DIM + kp, 0, 0);
            __builtin_prefetch(b + nrow * D_DIM + kp, 0, 0);
        }

        v8f c0 = {};
        v8f c1 = {};
        #pragma unroll
        for (int k = 0; k < D_DIM; k += 8) {
            float2 av0 = *(const float2*)(pa + k);
            float2 bv0 = *(const float2*)(pb + k);
            float2 av1 = *(const float2*)(pa + k + 4);
            float2 bv1 = *(const float2*)(pb + k + 4);
            v2f d0, d1;
            d0.x = av0.x - bv0.x;  d0.y = av0.y - bv0.y;
            d1.x = av1.x - bv1.x;  d1.y = av1.y - bv1.y;
            d0.x *= d0.x;  d0.y *= d0.y;
            d1.x *= d1.x;  d1.y *= d1.y;
            // D = A(diff^2) x B(ones) + C  ->  C[r][*] += rowsum chunk
            c0 = __builtin_amdgcn_wmma_f32_16x16x4_f32(
                    false, d0, false, ones, (short)0, c0, false, false);
            c1 = __builtin_amdgcn_wmma_f32_16x16x4_f32(
                    false, d1, false, ones, (short)0, c1, false, false);
        }
        v8f c = c0 + c1;   // lanes<16: c[v]=rowsum(v); lanes>=16: c[v]=rowsum(v+8)

        float partial = 0.0f;
        if (m == 0) {                       // lanes 0 and 16 finalize 8 rows each
            long row = rowbase + ((lane >> 4) << 3);
            // binary search: off[s] <= row < off[s+1]
            int lo = 0, hi = nseg - 1;
            while (lo < hi) {
                int mid = (lo + hi + 1) >> 1;
                if ((long)off[mid] <= row) lo = mid; else hi = mid - 1;
            }
            int s = lo;
            int segend = off[s + 1];
            #pragma unroll
            for (int v = 0; v < 8; ++v) {
                while (row >= (long)segend) { ++s; segend = off[s + 1]; }
                float len = (float)(segend - off[s]);
                partial += __fsqrt_rn(c[v]) / len;
                ++row;
            }
        }
        acc += partial;
    }

    // combine lane 0 (rows 0..7) and lane 16 (rows 8..15) partials, one atomic/wave
    acc += __shfl_xor(acc, 16, 32);
    if (lane == 0) atomicAdd(out, acc);
}

// ---------------------------------------------------------------------------
extern "C" void kernel_launch(void* const* d_in, const int* in_sizes, int n_in,
                              void* d_out, int out_size, void* d_ws, size_t ws_size,
                              hipStream_t stream)
{
    const float* a   = (const float*)d_in[0];   // clip_remap [N, 128] f32
    const float* b   = (const float*)d_in[1];   // clip_emb   [N, 128] f32
    const int*   num = (const int*)d_in[2];     // num_list   [nseg]   i32
    float* out = (float*)d_out;                 // scalar f32

    const int nseg   = in_sizes[2];
    const int nrows  = in_sizes[0] / D_DIM;
    const int ntiles = nrows / 16;

    int* off = (int*)d_ws;                      // nseg+1 ints (~8.2 KB)

    seg_scan_kernel<<<1, 256, 0, stream>>>(num, nseg, off, out);
    euclid_loss_kernel<<<2048, 256, 0, stream>>>(a, b, off, nseg, ntiles, out);
}